// EMLTreeHybrid_3891240370259
// MI455X (gfx1250) — compile-verified
//
#include <hip/hip_runtime.h>
#include <stdint.h>

// ---------------------------------------------------------------------------
// EML tree hybrid, MI455X (gfx1250).
// FP64 transcendental-bound (no f64 WMMA exists on CDNA5 -> matrix units
// inapplicable). CDNA5-specific paths: TDM tensor_load_to_lds (unguarded this
// round to prove emission), s_wait_tensorcnt, global_prefetch_b8, wave32
// shuffle butterfly. One wave per batch element -> 16384 waves (vs 512).
// ---------------------------------------------------------------------------

#define N_LEAVES   1024
#define N_SUB      1022
#define BATCH      16384
#define BLK        256
#define EXP_CLAMP  500.0
#define EML_CLAMP  1e300
#define BYPASS_THR 0.9999999999999998   // 1 - 2^-52

// workspace layout (doubles)
#define WS_LEAF  0           // 3072 doubles (leaf softmax probs)
#define WS_GATE  3072        // 2044 doubles (gate sigmoids)
#define WS_ROOT  5116        // 8 doubles (root softmax probs)
#define WS_TOTAL 5124

// output layout (doubles)
#define OUT_LEAF 32768
#define OUT_GATE 35840

// LDS layout (doubles) inside one shared array
#define GSLOT(n)   (2 * (n) + ((n) >> 4))     // monotone swizzle, max 2105
#define L_LEAF     0                          // 3072 (transposed leaf table)
#define L_GATE     3072                       // 2112 (swizzled gate pairs)
#define L_ROOT     5184                       // 8
#define L_RAW      5192                       // 5124 raw staging, reused as acc
#define L_TOTAL    10316                      // 82528 bytes

typedef unsigned int uint32x4 __attribute__((ext_vector_type(4)));
typedef int          int32x4  __attribute__((ext_vector_type(4)));
typedef int          int32x8  __attribute__((ext_vector_type(8)));

// ---------------------------------------------------------------------------
__device__ __forceinline__ double sanit(double v) {
    v = isnan(v) ? 0.0 : v;
    v = v >  EML_CLAMP ?  EML_CLAMP : v;
    v = v < -EML_CLAMP ? -EML_CLAMP : v;
    return v;
}

// eml(a,b) = safe_exp(a) - conj(log(conj(b))), sanitized
__device__ __forceinline__ void eml_san(double ar, double ai, double br, double bi,
                                        double* outr, double* outi) {
    double rc = ar;
    rc = rc < -EXP_CLAMP ? -EXP_CLAMP : rc;
    rc = rc >  EXP_CLAMP ?  EXP_CLAMP : rc;
    double e = exp(rc);
    double sn, cs;
    sincos(ai, &sn, &cs);
    double exr = e * cs;
    double exi = e * sn;
    double L = log(hypot(br, bi));          // Re log(conj b)
    double A = atan2(-bi, br);              // Im log(conj b); conj -> -A
    *outr = sanit(exr - L);
    *outi = sanit(exi + A);                 // exi - (-A)
}

// sigmoid blend of (left,right) then eml
__device__ __forceinline__ void combine(double lre, double lim, double rre, double rim,
                                        double sl, double sr, double* ore, double* oi) {
    double oml = 1.0 - sl, omr = 1.0 - sr;
    double ar, ai, br, bi;
    if (sl > BYPASS_THR) { ar = 1.0; ai = 0.0; }
    else                 { ar = sl + oml * lre; ai = oml * lim; }
    if (sr > BYPASS_THR) { br = 1.0; bi = 0.0; }
    else                 { br = sr + omr * rre; bi = omr * rim; }
    eml_san(ar, ai, br, bi, ore, oi);
}

// ---------------------------------------------------------------------------
// Kernel A: batch-invariant tables
// ---------------------------------------------------------------------------
__global__ void eml_prep(const double* __restrict__ leaf_logits,
                         const double* __restrict__ blend_logits,
                         const double* __restrict__ root_logits,
                         double* __restrict__ out_leaf,
                         double* __restrict__ out_gate,
                         double* __restrict__ ws) {
    int tid = blockIdx.x * blockDim.x + threadIdx.x;
    if (tid < N_LEAVES) {
        double a = leaf_logits[3 * tid];
        double b = leaf_logits[3 * tid + 1];
        double c = leaf_logits[3 * tid + 2];
        double m  = fmax(a, fmax(b, c));
        double ea = exp(a - m), eb = exp(b - m), ec = exp(c - m);
        double inv = 1.0 / (ea + eb + ec);
        double p0 = ea * inv, p1 = eb * inv, p2 = ec * inv;
        out_leaf[3 * tid]     = p0;  ws[WS_LEAF + 3 * tid]     = p0;
        out_leaf[3 * tid + 1] = p1;  ws[WS_LEAF + 3 * tid + 1] = p1;
        out_leaf[3 * tid + 2] = p2;  ws[WS_LEAF + 3 * tid + 2] = p2;
    }
    if (tid < N_SUB) {
        double z0 = blend_logits[2 * tid];
        double z1 = blend_logits[2 * tid + 1];
        double s0 = 1.0 / (1.0 + exp(-z0));
        double s1 = 1.0 / (1.0 + exp(-z1));
        out_gate[2 * tid]     = s0;  ws[WS_GATE + 2 * tid]     = s0;
        out_gate[2 * tid + 1] = s1;  ws[WS_GATE + 2 * tid + 1] = s1;
    }
    if (tid == 0) {
        for (int r = 0; r < 2; ++r) {
            double l0 = root_logits[4 * r], l1 = root_logits[4 * r + 1];
            double l2 = root_logits[4 * r + 2], l3 = root_logits[4 * r + 3];
            double m = fmax(fmax(l0, l1), fmax(l2, l3));
            double e0 = exp(l0 - m), e1 = exp(l1 - m), e2 = exp(l2 - m), e3 = exp(l3 - m);
            double inv = 1.0 / (e0 + e1 + e2 + e3);
            ws[WS_ROOT + 4 * r]     = e0 * inv;
            ws[WS_ROOT + 4 * r + 1] = e1 * inv;
            ws[WS_ROOT + 4 * r + 2] = e2 * inv;
            ws[WS_ROOT + 4 * r + 3] = e3 * inv;
        }
    }
}

// ---------------------------------------------------------------------------
// TDM: one 1-row 2D descriptor, data_size=8B (CDNA5 ISA sec 8.3/8.4).
// UNGUARDED: a clean compile proves tensor_load_to_lds lowered.
// ---------------------------------------------------------------------------
__device__ __forceinline__ void tdm_load_f64(const void* gptr, unsigned lds_byte_off,
                                             unsigned nelem) {
    unsigned long long ga = (unsigned long long)(uintptr_t)gptr;
    uint32x4 g0;
    g0.x = 1u;                                   // count=1, is_restore=0, gather=0
    g0.y = lds_byte_off;                         // lds_addr [63:32]
    g0.z = (unsigned)(ga & 0xFFFFFFFFu);         // global_addr [95:64]
    g0.w = (unsigned)((ga >> 32) & 0x01FFFFFFu)  // global_addr [120:96]
         | (2u << 30);                           // type=2
    int32x8 g1;
    g1[0] = (int)(3u << 16);                             // data_size=3 (8B)
    g1[1] = (int)((nelem & 0xFFFFu) << 16);              // tensor_dim0[15:0]
    g1[2] = (int)(((nelem >> 16) & 0xFFFFu) | (1u << 16)); // dim0 hi | tensor_dim1=1
    g1[3] = (int)((nelem & 0xFFFFu) << 16);              // tile_dim0 (16-bit)
    g1[4] = 1;                                           // tile_dim1=1
    g1[5] = (int)nelem;                                  // tensor_dim0_stride lo32
    g1[6] = 0;
    g1[7] = 0;
    int32x4 z4 = {0, 0, 0, 0};
#if defined(__clang_major__) && __clang_major__ >= 23
    int32x8 z8 = {0, 0, 0, 0, 0, 0, 0, 0};
    __builtin_amdgcn_tensor_load_to_lds(g0, g1, z4, z4, z8, 0);
#else
    __builtin_amdgcn_tensor_load_to_lds(g0, g1, z4, z4, 0);
#endif
}

// ---------------------------------------------------------------------------
// Kernel B: one wave32 per batch element. Lane L reduces contiguous leaves
// [32L, 32L+32) to its level-5 subtree value (pending left-siblings in LDS),
// then shfl_xor butterfly does levels 6..9, then redundant root eml.
// ---------------------------------------------------------------------------
__global__ __launch_bounds__(BLK) void eml_tree(const double* __restrict__ x,
                                                const double* __restrict__ y,
                                                const double* __restrict__ ws,
                                                double* __restrict__ out) {
    __shared__ double smem[L_TOTAL];             // 82528 B
    double* leafT = smem + L_LEAF;               // transposed leaf table
    double* gateS = smem + L_GATE;               // swizzled gate pairs
    double* rootp = smem + L_ROOT;
    double* rawD  = smem + L_RAW;                // staging, reused as acc stack

    const int tid  = threadIdx.x;
    const int lane = tid & 31;
    const int wv   = tid >> 5;
    const int elem = blockIdx.x * (BLK / 32) + wv;

    __builtin_prefetch(&x[elem], 0, 1);          // global_prefetch_b8
    __builtin_prefetch(&y[elem], 0, 1);

    if (tid == 0) {                              // TDM ignores EXEC; waves 1..7 skip via execz
        tdm_load_f64(ws, (unsigned)(uintptr_t)(&rawD[0]), WS_TOTAL);
        __builtin_amdgcn_s_wait_tensorcnt(0);
    }
    __syncthreads();

    // one-time swizzle pass: raw -> conflict-free layouts
    for (int i = tid; i < N_LEAVES; i += BLK) {
        double w0 = rawD[3 * i], w1 = rawD[3 * i + 1], w2 = rawD[3 * i + 2];
        int t = ((i & 31) * 32 + (i >> 5)) * 3;  // transpose 32x32 leaf grid
        leafT[t] = w0; leafT[t + 1] = w1; leafT[t + 2] = w2;
    }
    for (int n = tid; n < N_SUB; n += BLK) {
        int s = GSLOT(n);
        gateS[s]     = rawD[3072 + 2 * n];
        gateS[s + 1] = rawD[3072 + 2 * n + 1];
    }
    if (tid < 8) rootp[tid] = rawD[5116 + tid];
    __syncthreads();

    double* accR = rawD;                         // 5*BLK doubles
    double* accI = rawD + 5 * BLK;               // 5*BLK doubles (<= 5124 total)

    double xv = x[elem];
    double yv = y[elem];

    // ---- intra-lane: reduce 32 contiguous leaves to level-5 value ----
    double vre = 0.0, vim = 0.0;
    for (int j = 0; j < 32; ++j) {
        int g = (lane << 5) | j;                 // global leaf index
        int t = (j * 32 + lane) * 3;             // transposed slot: conflict-free
        vre = leafT[t] + leafT[t + 1] * xv + leafT[t + 2] * yv;
        vim = 0.0;
        bool stored = false;
        for (int lvl = 0; lvl < 5 && !stored; ++lvl) {
            if (((j >> lvl) & 1) == 0) {
                accR[lvl * BLK + tid] = vre;
                accI[lvl * BLK + tid] = vim;
                stored = true;
            } else {
                int node = (N_LEAVES - (N_LEAVES >> lvl)) + (g >> (lvl + 1));
                int gs = GSLOT(node);
                combine(accR[lvl * BLK + tid], accI[lvl * BLK + tid], vre, vim,
                        gateS[gs], gateS[gs + 1], &vre, &vim);
            }
        }
        // j==31 falls through with 5 merges done: vre/vim = level-5 value
    }

    // ---- cross-lane butterfly: levels 6..9 ----
#pragma unroll
    for (int s = 0; s < 4; ++s) {
        double pre = __shfl_xor(vre, 1 << s, 32);
        double pim = __shfl_xor(vim, 1 << s, 32);
        bool rgt = (lane >> s) & 1;
        double lre = rgt ? pre : vre, lim = rgt ? pim : vim;
        double rre = rgt ? vre : pre, rim = rgt ? vim : pim;
        int node = (N_LEAVES - (32 >> s)) + (lane >> (s + 1));
        int gs = GSLOT(node);
        combine(lre, lim, rre, rim, gateS[gs], gateS[gs + 1], &vre, &vim);
    }
    // lanes 0-15 hold cur[:,0] (left), 16-31 hold cur[:,1] (right)
    double ore = __shfl_xor(vre, 16, 32);
    double oim = __shfl_xor(vim, 16, 32);
    bool hi = lane >= 16;
    double c9lr = hi ? ore : vre, c9li = hi ? oim : vim;
    double c9rr = hi ? vre : ore, c9ri = hi ? vim : oim;

    // ---- root combine (all lanes redundant, lane 0 stores) ----
    double lire = rootp[0] + rootp[1] * xv + rootp[2] * yv + rootp[3] * c9lr;
    double liim = rootp[3] * c9li;
    double rire = rootp[4] + rootp[5] * xv + rootp[6] * yv + rootp[7] * c9rr;
    double riim = rootp[7] * c9ri;

    double rr, ri;
    eml_san(lire, liim, rire, riim, &rr, &ri);
    if (lane == 0) {
        out[2 * elem]     = rr;
        out[2 * elem + 1] = ri;
    }
}

// ---------------------------------------------------------------------------
extern "C" void kernel_launch(void* const* d_in, const int* in_sizes, int n_in,
                              void* d_out, int out_size, void* d_ws, size_t ws_size,
                              hipStream_t stream) {
    const double* x            = (const double*)d_in[0];
    const double* y            = (const double*)d_in[1];
    const double* leaf_logits  = (const double*)d_in[2];
    const double* blend_logits = (const double*)d_in[3];
    const double* root_logits  = (const double*)d_in[4];
    double* out = (double*)d_out;
    double* ws  = (double*)d_ws;

    eml_prep<<<4, 256, 0, stream>>>(leaf_logits, blend_logits, root_logits,
                                    out + OUT_LEAF, out + OUT_GATE, ws);
    // one wave32 per batch element: 16384 waves, 8 per block
    eml_tree<<<(BATCH * 32) / BLK, BLK, 0, stream>>>(x, y, ws, out);
}